// GraphLoss_55035710931515
// MI455X (gfx1250) — compile-verified
//
#include <hip/hip_runtime.h>
#include <stdint.h>

#define B_ 4
#define N_ 2048
#define G_ 1024
#define C_ 3
#define M_ (N_ + 1)

typedef __attribute__((ext_vector_type(2))) float v2f;
typedef __attribute__((ext_vector_type(8))) float v8f;

__device__ __forceinline__ unsigned long long shflxor_u64(unsigned long long v, int mask) {
    unsigned lo = (unsigned)v, hi = (unsigned)(v >> 32);
    lo = __shfl_xor(lo, mask, 32);
    hi = __shfl_xor(hi, mask, 32);
    return ((unsigned long long)hi << 32) | lo;
}

// ---------------------------------------------------------------- zero fill A
__global__ void k_zero(float* p, long total) {
    long tid = (long)blockIdx.x * blockDim.x + threadIdx.x;
    long stride = (long)gridDim.x * blockDim.x;
    uintptr_t a = (uintptr_t)p;
    long head = (long)(((16u - (unsigned)(a & 15u)) & 15u) >> 2);
    if (head > total) head = total;
    for (long i = tid; i < head; i += stride) p[i] = 0.0f;
    float4* v = (float4*)(p + head);
    long nv = (total - head) >> 2;
    float4 z; z.x = z.y = z.z = z.w = 0.0f;
    for (long i = tid; i < nv; i += stride) v[i] = z;   // global_store_b128
    long done = head + (nv << 2);
    for (long i = done + tid; i < total; i += stride) p[i] = 0.0f;
}

// ---------------------------------------------------------------- init scratch
__global__ void k_init(unsigned long long* best_key, unsigned long long* keep_key,
                       int* col_cnt, int* min_row, int* nmax, float* acc) {
    int tid = blockIdx.x * blockDim.x + threadIdx.x;
    if (tid < B_ * G_) best_key[tid] = ~0ull;
    if (tid < B_ * N_) { keep_key[tid] = ~0ull; col_cnt[tid] = 0; min_row[tid] = 0x7fffffff; }
    if (tid < B_) nmax[tid] = 0;
    if (tid < 4) acc[tid] = 0.0f;
}

// ------------------------------------------------- WMMA nearest / best_pred_g
// One wave handles a 16-row strip; each V_WMMA_F32_16X16X4_F32 produces a
// 16x16 tile of exact squared distances via the rank-4 trick:
//   A row = [x, y, x^2+y^2, 1],  B col = [-2bx, -2by, 1, bx^2+by^2].
// The B-operand table is staged once per workgroup in LDS (conflict-free
// ds_load_b64 per tile), and the g-loop is unrolled x2 so two WMMAs overlap.
__global__ void __launch_bounds__(256) k_nearest(
        const float* __restrict__ positions,
        const float* __restrict__ gt_pts,
        const int*   __restrict__ gt_type,
        const float* __restrict__ semantics,
        float* __restrict__ out_semgt,
        int* __restrict__ nearest,
        unsigned long long* __restrict__ best_key,
        int* __restrict__ nmax, float* __restrict__ acc) {
    __shared__ float4 bstage[G_];            // 16 KB of 320 KB/WGP

    const int lane = threadIdx.x & 31;
    const int wave = threadIdx.x >> 5;
    const int strip = blockIdx.x * (blockDim.x >> 5) + wave;   // 0 .. B*(N/16)-1
    const int b  = strip / (N_ / 16);        // uniform across the block (8 | 128)
    const int m0 = (strip % (N_ / 16)) * 16;
    const int half = lane >> 4;
    const int sub  = lane & 15;

    // cooperative stage of the B-operand table for this batch
    for (int g = threadIdx.x; g < G_; g += blockDim.x) {
        float bx = gt_pts[((long)b * G_ + g) * 2 + 0];
        float by = gt_pts[((long)b * G_ + g) * 2 + 1];
        float4 t;
        t.x = -2.0f * bx; t.y = -2.0f * by; t.z = 1.0f; t.w = bx * bx + by * by;
        bstage[g] = t;
    }
    __syncthreads();

    // A operand: lanes 0-15 hold (K0,K1)=(x,y); lanes 16-31 hold (K2,K3)=(r2,1)
    const int ma = m0 + sub;
    float px = positions[((long)b * N_ + ma) * 2 + 0] * 0.15f + (-29.925f);
    float py = positions[((long)b * N_ + ma) * 2 + 1] * 0.15f + (-14.925f);
    float r2 = px * px + py * py;
    v2f aop;
    aop.x = half ? r2   : px;
    aop.y = half ? 1.0f : py;

    unsigned long long best[8];
#pragma unroll
    for (int r = 0; r < 8; ++r) best[r] = ~0ull;

    // lane's LDS view: two v2f halves per column
    const float2* bs = (const float2*)bstage;

#pragma unroll 2
    for (int g0 = 0; g0 < G_; g0 += 16) {
        const int gc = g0 + sub;
        float2 bb = bs[gc * 2 + half];       // ds_load_b64, conflict-free
        v2f bop; bop.x = bb.x; bop.y = bb.y;
        v8f cz = {};
        v8f d = __builtin_amdgcn_wmma_f32_16x16x4_f32(
            /*neg_a=*/false, aop, /*neg_b=*/false, bop,
            /*c_mod=*/(short)0, cz, /*reuse_a=*/false, /*reuse_b=*/false);
#pragma unroll
        for (int r = 0; r < 8; ++r) {
            float d2 = fmaxf(d[r], 0.0f);    // cancellation guard, single v_max
            unsigned long long key =
                ((unsigned long long)__float_as_uint(d2) << 32) | (unsigned)gc;
            if (key < best[r]) best[r] = key;
        }
    }
    // argmin across the 16 lanes sharing each row (xor<16 stays inside half)
#pragma unroll
    for (int r = 0; r < 8; ++r) {
        unsigned long long k = best[r];
        for (int ofs = 1; ofs < 16; ofs <<= 1) {
            unsigned long long o = shflxor_u64(k, ofs);
            if (o < k) k = o;
        }
        best[r] = k;
    }
    if (sub < 8) {   // writer lanes: 0-7 -> rows 0-7, 16-23 -> rows 8-15
        const int m = m0 + sub + 8 * half;
        unsigned long long k = best[sub];
        int   g  = (int)(unsigned)(k & 0xffffffffull);
        float d2 = __uint_as_float((unsigned)(k >> 32));
        float cmin = sqrtf(d2);
        nearest[b * N_ + m] = g;
        atomicMax(&nmax[b], g);
        atomicMin(&best_key[b * G_ + g],
                  ((unsigned long long)__float_as_uint(d2) << 32) | (unsigned)m);
        int cls = gt_type[b * G_ + g];
#pragma unroll
        for (int c = 0; c < C_; ++c)
            out_semgt[((long)b * C_ + c) * N_ + m] = (c == cls) ? 1.0f : 0.0f;
        atomicAdd(&acc[0], cmin);
        atomicAdd(&acc[1], semantics[((long)b * C_ + cls) * N_ + m]);
    }
}

// ---------------------------------------------------------------- walk loop
__global__ void k_walk(const int* __restrict__ nearest, const int* __restrict__ nmax,
                       const int* __restrict__ gt_ins,
                       const unsigned long long* __restrict__ best_key,
                       int* __restrict__ next_pred, int* __restrict__ col_cnt,
                       int* __restrict__ min_row) {
    int tid = blockIdx.x * blockDim.x + threadIdx.x;
    if (tid >= B_ * N_) return;
    const int b = tid / N_, n = tid % N_;
    const int* ins = gt_ins + b * G_;
    const unsigned long long* bk = best_key + b * G_;
    const int nm = nmax[b];
    int idx = nearest[tid];
    int steps = 0, res = -1;
    while (true) {
        bool is_end = idx >= nm;
        int g_next_raw = is_end ? -1 : idx + 1;
        int g_next = ((g_next_raw % G_) + G_) % G_;
        bool ex = !is_end && (bk[g_next] != ~0ull);
        int idxm = ((idx % G_) + G_) % G_;
        bool same = ins[idxm] == ins[g_next];
        bool matched = ex && same;
        bool stop = matched || !same || (steps >= G_ + 2);
        res = matched ? (int)(unsigned)(bk[g_next] & 0xffffffffull) : -1;
        idx = g_next_raw;
        ++steps;
        if (stop) break;
    }
    next_pred[tid] = res;
    if (res >= 0) {
        atomicAdd(&col_cnt[b * N_ + res], 1);
        atomicMin(&min_row[b * N_ + res], n);
    }
}

// --------------------------------------- column dedup: nearest-in-dist keeper
__global__ void k_keep(const float* __restrict__ positions,
                       const int* __restrict__ next_pred, const int* __restrict__ col_cnt,
                       unsigned long long* __restrict__ keep_key) {
    int tid = blockIdx.x * blockDim.x + threadIdx.x;
    if (tid >= B_ * N_) return;
    const int b = tid / N_, m = tid % N_;
    const int c = next_pred[tid];
    if (c < 0 || col_cnt[b * N_ + c] <= 1) return;
    float mx = positions[((long)b * N_ + m) * 2 + 0] * 0.15f + (-29.925f);
    float my = positions[((long)b * N_ + m) * 2 + 1] * 0.15f + (-14.925f);
    float cx = positions[((long)b * N_ + c) * 2 + 0] * 0.15f + (-29.925f);
    float cy = positions[((long)b * N_ + c) * 2 + 1] * 0.15f + (-14.925f);
    float dx = mx - cx, dy = my - cy;
    float d2 = fmaxf(dx * dx + dy * dy, 0.0f);
    atomicMin(&keep_key[b * N_ + c],
              ((unsigned long long)__float_as_uint(d2) << 32) | (unsigned)m);
}

// ------------------------------------------------- row scatter + forward loss
__global__ void k_rows(const float* __restrict__ matches,
                       const int* __restrict__ next_pred, const int* __restrict__ col_cnt,
                       const unsigned long long* __restrict__ keep_key,
                       float* __restrict__ outA, float* __restrict__ acc) {
    int tid = blockIdx.x * blockDim.x + threadIdx.x;
    if (tid >= B_ * N_) return;
    const int b = tid / N_, m = tid % N_;
    const int c = next_pred[tid];
    bool kept = false;
    if (c >= 0) {
        int cnt = col_cnt[b * N_ + c];
        kept = (cnt <= 1) ||
               ((int)(unsigned)(keep_key[b * N_ + c] & 0xffffffffull) == m);
    }
    float* A = outA + (long)b * M_ * M_;
    int tgt_f;
    if (kept) { A[(long)m * M_ + c] = 1.0f;        tgt_f = c; }
    else      { A[(long)m * M_ + (M_ - 1)] = 1.0f; tgt_f = M_ - 1; }
    atomicAdd(&acc[2], matches[(long)b * M_ * M_ + (long)m * M_ + tgt_f]);
}

// ------------------------------------------------ column pass + backward loss
__global__ void k_cols(const float* __restrict__ matches,
                       const int* __restrict__ col_cnt, const int* __restrict__ min_row,
                       const unsigned long long* __restrict__ keep_key,
                       float* __restrict__ outA, float* __restrict__ acc) {
    int tid = blockIdx.x * blockDim.x + threadIdx.x;
    if (tid >= B_ * N_) return;
    const int b = tid / N_, c = tid % N_;
    const int cnt = col_cnt[tid];
    float* A = outA + (long)b * M_ * M_;
    int tgt_b;
    if (cnt == 0)      { A[(long)(M_ - 1) * M_ + c] = 1.0f; tgt_b = M_ - 1; }
    else if (cnt == 1) { tgt_b = min_row[tid]; }
    else               { tgt_b = (int)(unsigned)(keep_key[tid] & 0xffffffffull); }
    atomicAdd(&acc[3], matches[(long)b * M_ * M_ + (long)c * M_ + tgt_b]);
}

// ---------------------------------------------------------------- finalize
__global__ void k_final(const float* __restrict__ acc, float* __restrict__ out) {
    out[0] =  acc[0] / (float)(B_ * N_);
    out[1] = -0.5f * (acc[2] + acc[3]) / (float)(B_ * (M_ - 1));
    out[2] = -acc[1] / (float)(B_ * N_);
}

extern "C" void kernel_launch(void* const* d_in, const int* in_sizes, int n_in,
                              void* d_out, int out_size, void* d_ws, size_t ws_size,
                              hipStream_t stream) {
    const float* matches   = (const float*)d_in[0];
    const float* positions = (const float*)d_in[1];
    const float* semantics = (const float*)d_in[2];
    /* d_in[3] = masks (unused by reference) */
    const float* gt_pts    = (const float*)d_in[4];
    const int*   gt_ins    = (const int*)d_in[5];
    const int*   gt_type   = (const int*)d_in[6];
    float* out = (float*)d_out;

    char* ws = (char*)d_ws;
    unsigned long long* best_key = (unsigned long long*)ws; ws += sizeof(unsigned long long) * B_ * G_;
    unsigned long long* keep_key = (unsigned long long*)ws; ws += sizeof(unsigned long long) * B_ * N_;
    int*   nearest   = (int*)ws;   ws += sizeof(int) * B_ * N_;
    int*   next_pred = (int*)ws;   ws += sizeof(int) * B_ * N_;
    int*   col_cnt   = (int*)ws;   ws += sizeof(int) * B_ * N_;
    int*   min_row   = (int*)ws;   ws += sizeof(int) * B_ * N_;
    int*   nmax      = (int*)ws;   ws += sizeof(int) * B_;
    float* acc       = (float*)ws; ws += sizeof(float) * 4;
    (void)ws_size; (void)in_sizes; (void)n_in; (void)out_size;

    float* outA = out + 3;
    float* outS = out + 3 + (long)B_ * M_ * M_;
    const long totalA = (long)B_ * M_ * M_;

    k_zero<<<2048, 256, 0, stream>>>(outA, totalA);
    k_init<<<(B_ * N_ + 255) / 256, 256, 0, stream>>>(best_key, keep_key, col_cnt, min_row, nmax, acc);
    k_nearest<<<(B_ * (N_ / 16)) / 8, 256, 0, stream>>>(positions, gt_pts, gt_type, semantics,
                                                        outS, nearest, best_key, nmax, acc);
    k_walk<<<(B_ * N_ + 255) / 256, 256, 0, stream>>>(nearest, nmax, gt_ins, best_key,
                                                      next_pred, col_cnt, min_row);
    k_keep<<<(B_ * N_ + 255) / 256, 256, 0, stream>>>(positions, next_pred, col_cnt, keep_key);
    k_rows<<<(B_ * N_ + 255) / 256, 256, 0, stream>>>(matches, next_pred, col_cnt, keep_key, outA, acc);
    k_cols<<<(B_ * N_ + 255) / 256, 256, 0, stream>>>(matches, col_cnt, min_row, keep_key, outA, acc);
    k_final<<<1, 1, 0, stream>>>(acc, out);
}